// MemSE_45449343927014
// MI455X (gfx1250) — compile-verified
//
#include <hip/hip_runtime.h>

typedef __attribute__((ext_vector_type(2))) float v2f;
typedef __attribute__((ext_vector_type(8))) float v8f;

#define SIGMA 0.01f
#define NB   4
#define DIN  3072
#define DH   2048
#define DOUT 1024

// ---- workspace float-offset layout ----
#define WS_XB      0                       // NB*DIN
#define WS_SUMX2   (WS_XB + NB*DIN)        // NB
#define WS_MU1     (WS_SUMX2 + NB)         // NB*DH
#define WS_MUSP    (WS_MU1 + NB*DH)        // NB*DH
#define WS_DVEC    (WS_MUSP + NB*DH)       // NB*DH
#define WS_VAR1    (WS_DVEC + NB*DH)       // NB
#define WS_VAR2    (WS_VAR1 + NB)          // NB
#define WS_P1      (WS_VAR2 + NB)          // NB
#define WS_ACC     (WS_P1 + NB)            // ACC_N accumulators
// accumulator slots (per batch, 4 each)
#define A_EMEM1 0
#define A_TP1   4
#define A_TN1   8
#define A_EMEM2 12
#define A_TP2   16
#define A_TN2   20
#define A_QP2   24
#define A_QN2   28
#define A_SUM2  32
#define ACC_N   36

__device__ __forceinline__ float block_reduce(float v, float* sm) {
    int t = threadIdx.x;
    sm[t] = v;
    __syncthreads();
    for (int s = 128; s > 0; s >>= 1) {
        if (t < s) sm[t] += sm[t + s];
        __syncthreads();
    }
    float r = sm[0];
    __syncthreads();
    return r;
}

// zero accumulators each launch (graph-replay safe)
__global__ void k_init(float* acc) {
    int t = threadIdx.x;
    if (t < ACC_N) acc[t] = 0.0f;
}

// xb = x + bias ; sumx2[b] = sum_j xb^2
__global__ void k_stage0(const float* __restrict__ x, const float* __restrict__ bias,
                         float* __restrict__ ws) {
    int b = blockIdx.x, t = threadIdx.x;
    __shared__ float sm[256];
    float s = 0.0f;
    for (int j = t; j < DIN; j += 256) {
        float v = x[b * DIN + j] + bias[j];
        ws[WS_XB + b * DIN + j] = v;
        s = fmaf(v, v, s);
    }
    float tot = block_reduce(s, sm);
    if (t == 0) ws[WS_SUMX2 + b] = tot;
}

// one block per row i of W1: mu1, (Gp@x)^2, (Gn@x)^2, e_mem partials
__global__ void k_l1rows(const float* __restrict__ W1, float* __restrict__ ws) {
    int i = blockIdx.x, t = threadIdx.x;
    __shared__ float sm[256];
    const float* xb = ws + WS_XB;
    float m[NB] = {}, p[NB] = {}, n[NB] = {}, em[NB] = {};
    for (int j = t; j < DIN; j += 256) {
        float w  = W1[(size_t)i * DIN + j];
        float wp = fmaxf(w, 0.0f), wn = fmaxf(-w, 0.0f), aw = wp + wn;
#pragma unroll
        for (int b = 0; b < NB; ++b) {
            float xv = xb[b * DIN + j];
            m[b]  = fmaf(w,  xv, m[b]);
            p[b]  = fmaf(wp, xv, p[b]);
            n[b]  = fmaf(wn, xv, n[b]);
            em[b] = fmaf(aw, xv * xv, em[b]);
        }
    }
    float* acc = ws + WS_ACC;
#pragma unroll
    for (int b = 0; b < NB; ++b) {
        float M = block_reduce(m[b],  sm);
        float P = block_reduce(p[b],  sm);
        float N = block_reduce(n[b],  sm);
        float E = block_reduce(em[b], sm);
        if (t == 0) {
            ws[WS_MU1 + b * DH + i] = M;
            atomicAdd(&acc[A_TP1 + b],   P * P);
            atomicAdd(&acc[A_TN1 + b],   N * N);
            atomicAdd(&acc[A_EMEM1 + b], E);
        }
    }
}

// var1[b], P1[b]
__global__ void k_scalars1(const float* __restrict__ Gmax1, const float* __restrict__ Wmax1,
                           float* __restrict__ ws) {
    int t = threadIdx.x;
    if (t < NB) {
        float c    = Gmax1[0] / Wmax1[0];
        float s2   = (SIGMA / c) * (SIGMA / c);
        float sx   = ws[WS_SUMX2 + t];
        float varp = s2 * sx;                 // (sigma/sqrt2 / c)^2 * 2 * sum(x^2)
        ws[WS_VAR1 + t] = 2.0f * varp;        // 2*(sigma/c)^2 * sum(x^2)
        float* acc = ws + WS_ACC;
        // P1 = c*emem + c*(tp + DH*varp) + c*(tn + DH*varp), r=1
        ws[WS_P1 + t] = c * (acc[A_EMEM1 + t] + acc[A_TP1 + t] + acc[A_TN1 + t]
                             + 2.0f * (float)DH * varp);
    }
}

// softplus moment propagation; gamma stays diagonal: d = var1*s^2
__global__ void k_softplus(float* __restrict__ ws) {
    int b = blockIdx.x, t = threadIdx.x;
    __shared__ float sm[256];
    float v1 = ws[WS_VAR1 + b];
    float s2sum = 0.0f;
    for (int j = t; j < DH; j += 256) {
        float mv = ws[WS_MU1 + b * DH + j];
        float s  = 1.0f / (1.0f + expf(-mv));
        float sp = fmaxf(mv, 0.0f) + log1pf(expf(-fabsf(mv)));
        float mu = sp + 0.5f * s * (1.0f - s) * v1;
        float d  = v1 * s * s;
        ws[WS_MUSP + b * DH + j] = mu;
        ws[WS_DVEC + b * DH + j] = d;
        s2sum += fmaf(mu, mu, d);
    }
    float tot = block_reduce(s2sum, sm);
    if (t == 0) ws[WS_ACC + A_SUM2 + b] = tot;
}

// one block per row i of W2: mu2 (to d_out), power partials for layer 2
__global__ void k_l2rows(const float* __restrict__ W2, float* __restrict__ ws,
                         float* __restrict__ out_mu) {
    int i = blockIdx.x, t = threadIdx.x;
    __shared__ float sm[256];
    const float* ms_ = ws + WS_MUSP;
    const float* dv  = ws + WS_DVEC;
    float m[NB] = {}, p[NB] = {}, n[NB] = {}, qp[NB] = {}, qn[NB] = {}, em[NB] = {};
    for (int j = t; j < DH; j += 256) {
        float w   = W2[(size_t)i * DH + j];
        float wp  = fmaxf(w, 0.0f), wn = fmaxf(-w, 0.0f), aw = wp + wn;
        float wp2 = wp * wp, wn2 = wn * wn;
#pragma unroll
        for (int b = 0; b < NB; ++b) {
            float mu = ms_[b * DH + j];
            float d  = dv[b * DH + j];
            m[b]  = fmaf(w,   mu, m[b]);
            p[b]  = fmaf(wp,  mu, p[b]);
            n[b]  = fmaf(wn,  mu, n[b]);
            qp[b] = fmaf(wp2, d,  qp[b]);
            qn[b] = fmaf(wn2, d,  qn[b]);
            em[b] = fmaf(aw, fmaf(mu, mu, d), em[b]);
        }
    }
    float* acc = ws + WS_ACC;
#pragma unroll
    for (int b = 0; b < NB; ++b) {
        float M  = block_reduce(m[b],  sm);
        float P  = block_reduce(p[b],  sm);
        float N  = block_reduce(n[b],  sm);
        float QP = block_reduce(qp[b], sm);
        float QN = block_reduce(qn[b], sm);
        float E  = block_reduce(em[b], sm);
        if (t == 0) {
            out_mu[b * DOUT + i] = M;
            atomicAdd(&acc[A_TP2 + b],   P * P);
            atomicAdd(&acc[A_TN2 + b],   N * N);
            atomicAdd(&acc[A_QP2 + b],   QP);
            atomicAdd(&acc[A_QN2 + b],   QN);
            atomicAdd(&acc[A_EMEM2 + b], E);
        }
    }
}

// var2[b], total power P[b] -> d_out
__global__ void k_scalars2(const float* __restrict__ Gmax2, const float* __restrict__ Wmax2,
                           float* __restrict__ ws, float* __restrict__ out_P) {
    int t = threadIdx.x;
    if (t < NB) {
        float c    = Gmax2[0] / Wmax2[0];
        float s2   = (SIGMA / c) * (SIGMA / c);
        float* acc = ws + WS_ACC;
        float sum2 = acc[A_SUM2 + t];
        float varp = s2 * sum2;
        ws[WS_VAR2 + t] = 2.0f * varp;
        float P2 = c * (acc[A_EMEM2 + t] + acc[A_TP2 + t] + acc[A_TN2 + t]
                        + acc[A_QP2 + t] + acc[A_QN2 + t] + 2.0f * (float)DOUT * varp);
        out_P[t] = ws[WS_P1 + t] + P2;
    }
}

// gamma2[b] = W2 * diag(d_b) * W2^T + var2[b]*I, fp32 WMMA 16x16x4.
// Wave computes a 64x32 tile: 4 row-fragments (unscaled) x 2 col-fragments
// (scaled by d) -> 8 WMMAs per 6 b64 loads per K=4 step (~10.7 FLOP/B).
// Block = 8 waves arranged 2x4 -> 128x128 block tile.
// A/B fragment lane layout (both are row tiles of W2):
//   lane = idx + 16*(k>=2), vgpr = k&1.
__global__ void k_gamma(const float* __restrict__ W2, const float* __restrict__ dvec,
                        const float* __restrict__ var2, float* __restrict__ out_gamma) {
    const int b    = blockIdx.z;
    const int lane = threadIdx.x & 31;
    const int wv   = threadIdx.x >> 5;                 // 0..7
    const int row0 = blockIdx.x * 128 + (wv >> 2) * 64; // 2 row groups of 64
    const int col0 = blockIdx.y * 128 + (wv & 3) * 32;  // 4 col groups of 32
    const int lm   = lane & 15;   // m (A) / n (B) within 16-tile
    const int kh   = lane >> 4;   // k half: selects k offset 0 or 2

    __shared__ float dsh[DH];
    for (int j = threadIdx.x; j < DH; j += 256) dsh[j] = dvec[b * DH + j];
    __syncthreads();

    v8f acc[4][2];
#pragma unroll
    for (int r = 0; r < 4; ++r)
#pragma unroll
        for (int c = 0; c < 2; ++c)
            acc[r][c] = (v8f){0, 0, 0, 0, 0, 0, 0, 0};

    const float* Ar[4];
#pragma unroll
    for (int r = 0; r < 4; ++r) Ar[r] = W2 + (size_t)(row0 + 16 * r + lm) * DH;
    const float* Bc[2];
#pragma unroll
    for (int c = 0; c < 2; ++c) Bc[c] = W2 + (size_t)(col0 + 16 * c + lm) * DH;

    for (int k = 0; k < DH; k += 4) {
        const int kk = k + 2 * kh;
        v2f xa[4];
#pragma unroll
        for (int r = 0; r < 4; ++r) xa[r] = *(const v2f*)(Ar[r] + kk);
        v2f yb[2];
#pragma unroll
        for (int c = 0; c < 2; ++c) yb[c] = *(const v2f*)(Bc[c] + kk);
        const float d0 = dsh[kk], d1 = dsh[kk + 1];
#pragma unroll
        for (int c = 0; c < 2; ++c) { yb[c][0] *= d0; yb[c][1] *= d1; }
#pragma unroll
        for (int r = 0; r < 4; ++r) {
#pragma unroll
            for (int c = 0; c < 2; ++c) {
                acc[r][c] = __builtin_amdgcn_wmma_f32_16x16x4_f32(
                    false, xa[r], false, yb[c], (short)0, acc[r][c], false, false);
            }
        }
    }

    const float vb = var2[b];
    float* G = out_gamma + (size_t)b * DOUT * DOUT;
    // C layout: lanes 0-15: N=lane, M=vgpr; lanes 16-31: N=lane-16, M=vgpr+8
    const int mb = 8 * kh;
#pragma unroll
    for (int r = 0; r < 4; ++r) {
#pragma unroll
        for (int c = 0; c < 2; ++c) {
            const int colg = col0 + 16 * c + lm;
#pragma unroll
            for (int v = 0; v < 8; ++v) {
                const int rowg = row0 + 16 * r + mb + v;
                float e = acc[r][c][v];
                if (rowg == colg) e += vb;
                G[(size_t)rowg * DOUT + colg] = e;
            }
        }
    }
}

extern "C" void kernel_launch(void* const* d_in, const int* in_sizes, int n_in,
                              void* d_out, int out_size, void* d_ws, size_t ws_size,
                              hipStream_t stream) {
    (void)in_sizes; (void)n_in; (void)out_size; (void)ws_size;
    const float* x     = (const float*)d_in[0];
    const float* bias  = (const float*)d_in[1];
    const float* W1    = (const float*)d_in[2];
    const float* W2    = (const float*)d_in[3];
    const float* Gmax1 = (const float*)d_in[4];
    const float* Gmax2 = (const float*)d_in[5];
    const float* Wmax1 = (const float*)d_in[6];
    const float* Wmax2 = (const float*)d_in[7];

    float* out       = (float*)d_out;
    float* ws        = (float*)d_ws;
    float* out_mu    = out;                                         // [NB, DOUT]
    float* out_gamma = out + NB * DOUT;                             // [NB, DOUT, DOUT]
    float* out_P     = out + NB * DOUT + (size_t)NB * DOUT * DOUT;  // [NB]

    k_init    <<<1, 64, 0, stream>>>(ws + WS_ACC);
    k_stage0  <<<NB, 256, 0, stream>>>(x, bias, ws);
    k_l1rows  <<<DH, 256, 0, stream>>>(W1, ws);
    k_scalars1<<<1, 32, 0, stream>>>(Gmax1, Wmax1, ws);
    k_softplus<<<NB, 256, 0, stream>>>(ws);
    k_l2rows  <<<DOUT, 256, 0, stream>>>(W2, ws, out_mu);
    k_scalars2<<<1, 32, 0, stream>>>(Gmax2, Wmax2, ws, out_P);

    dim3 g(DOUT / 128, DOUT / 128, NB);
    k_gamma   <<<g, 256, 0, stream>>>(W2, ws + WS_DVEC, ws + WS_VAR2, out_gamma);
}